// encoder_v2_10462540333218
// MI455X (gfx1250) — compile-verified
//
#include <hip/hip_runtime.h>
#include <hip/hip_bf16.h>
#include <math.h>
#include <stdint.h>

typedef __attribute__((ext_vector_type(2))) float v2f;
typedef __attribute__((ext_vector_type(8))) float v8f;

#define CCH   128          // channels
#define HW    131072       // 256*512
#define KS    9
#define LTOT  163840       // Q*M*M
#define MM    16384        // M*M
#define LTILE 16
#define NPAIR (LTILE*KS)   // 144
#define XSTR  132          // padded LDS row stride: 16B-aligned rows, spread banks

#define USE_ASYNC_GATHER 1

// sigmoid with hardware v_exp_f32 + v_rcp_f32 (no IEEE divide expansion)
__device__ __forceinline__ float fast_sigmoid(float v) {
    return __builtin_amdgcn_rcpf(1.0f + __expf(-v));
}

// ---------------- kernel 0: zero the stats accumulators ----------------
__global__ void k_init(float* ws) {
    int i = blockIdx.x * blockDim.x + threadIdx.x;
    if (i < 256) ws[i] = 0.0f;
}

// ---------------- kernel 1: gather + WMMA attention + depthwise dot ----
__global__ __launch_bounds__(128) void k_main(
        const float* __restrict__ x, const int* __restrict__ codes,
        const float* __restrict__ att_w, const float* __restrict__ att_b,
        const float* __restrict__ depth_w, const float* __restrict__ depth_b,
        float* __restrict__ out, float* __restrict__ stat_sum,
        float* __restrict__ stat_sq) {
    __shared__ float xe[NPAIR][XSTR];      // gathered tile  [l*9+k][c]
    __shared__ float feat[LTILE][20];      // avg(9) | max(9) | zero-pad(2)
    __shared__ float attl[LTILE][16];      // 1 + sigmoid(att)   [l][o]
    __shared__ float aw[16][20];           // zero-padded att_w (A matrix)

    const int tid = threadIdx.x;           // == channel index
    const float* xc = x + (size_t)tid * HW;

    // per-channel depthwise weights, loaded once
    float dw[KS];
#pragma unroll
    for (int k = 0; k < KS; ++k) dw[k] = depth_w[tid * KS + k];
    const float db = depth_b[tid];

    // ---- stage zero-padded A matrix (att_w: 9x18 -> 16x20) in LDS, once
    for (int i = tid; i < 16 * 20; i += CCH) {
        int o = i / 20, j = i - o * 20;
        aw[o][j] = (o < KS && j < 18) ? att_w[o * 18 + j] : 0.0f;
    }
    __syncthreads();

    // ---- hoist WMMA A-fragments + bias into registers (loop-invariant)
    const int half = (tid >> 4) & 1;       // K sub-rows 0,1 vs 2,3 per tile
    const int lm   = tid & 15;             // A: row(o)  B: col(l)  D: col(l)
    v2f afrag[5];
    float abias[8];
#pragma unroll
    for (int kt = 0; kt < 5; ++kt) {
        int j0 = 4 * kt + 2 * half;
        afrag[kt].x = aw[lm][j0];
        afrag[kt].y = aw[lm][j0 + 1];
    }
#pragma unroll
    for (int r = 0; r < 8; ++r) {
        int o = r + 8 * half;
        abias[r] = (o < KS) ? att_b[o] : 0.0f;
    }

    float lsum = 0.0f, lsq = 0.0f;
    const int ntiles = LTOT / LTILE;       // 10240
    const unsigned lds_col = (unsigned)(uintptr_t)(&xe[0][tid]);

    for (int tile = blockIdx.x; tile < ntiles; tile += gridDim.x) {
        const int l0 = tile * LTILE;
        const int* cb = codes + (size_t)l0 * KS;
        if (tid < LTILE * 2) feat[tid >> 1][18 + (tid & 1)] = 0.0f; // K pad

        // ---- phase 1a: random gather of 144 x-columns straight into LDS.
        // x (64MB) is L2-resident on MI455X (192MB L2); indices are wave-
        // uniform (scalar loads); data path is the CDNA5 async-to-LDS DMA.
#if USE_ASYNC_GATHER
        unsigned lds_addr = lds_col;
#pragma unroll 4
        for (int p = 0; p < NPAIR; ++p) {
            int idx = cb[p];
            unsigned long long ga = (unsigned long long)(uintptr_t)(xc + idx);
            asm volatile("global_load_async_to_lds_b32 %0, %1, off"
                         :: "v"(lds_addr), "v"(ga) : "memory");
            lds_addr += XSTR * 4;
        }
        asm volatile("s_wait_asynccnt 0x0" ::: "memory");
#else
#pragma unroll 4
        for (int p = 0; p < NPAIR; ++p)
            xe[p][tid] = xc[cb[p]];
#endif
        __syncthreads();

        // ---- phase 1b: channel reductions -> feat (avg | max), b128 reads
        for (int p = tid; p < NPAIR; p += CCH) {
            const float4* row = (const float4*)(&xe[p][0]);
            float s = 0.0f, m = -INFINITY;
#pragma unroll 4
            for (int c4 = 0; c4 < CCH / 4; ++c4) {
                float4 v = row[c4];
                s += v.x + v.y + v.z + v.w;
                m = fmaxf(m, fmaxf(fmaxf(v.x, v.y), fmaxf(v.z, v.w)));
            }
            int li = p / KS, k = p - li * KS;
            feat[li][k]      = s * (1.0f / CCH);
            feat[li][KS + k] = m;
        }
        __syncthreads();

        // ---- phase 2: wave 0: att = sigmoid(A(9x18) * feat(18x16) + b)
        // V_WMMA_F32_16X16X4_F32, K padded 18 -> 20 (5 K-tiles)
        if (tid < 32) {
            v8f acc = {};
#pragma unroll
            for (int kt = 0; kt < 5; ++kt) {
                int j0 = 4 * kt + 2 * half;
                v2f b;
                b.x = feat[lm][j0];
                b.y = feat[lm][j0 + 1];
                acc = __builtin_amdgcn_wmma_f32_16x16x4_f32(
                        false, afrag[kt], false, b, (short)0, acc, false, false);
            }
            // D layout: vgpr r, lanes<16 -> o=r, lanes>=16 -> o=r+8
            // store 1 + sigmoid so phase 3 is pure mul+fma
#pragma unroll
            for (int r = 0; r < 8; ++r)
                attl[lm][r + 8 * half] =
                        1.0f + fast_sigmoid(acc[r] + abias[r]);
        }
        __syncthreads();

        // ---- phase 3: depthwise dot + vectorized scatter + stats
        const int q  = l0 / MM;
        const int r0 = l0 - q * MM;        // tile never crosses MM boundary
        float* obase = out + (size_t)q * CCH * MM + (size_t)tid * MM + r0;
        float accs[LTILE];
#pragma unroll
        for (int li = 0; li < LTILE; ++li) {
            float acc = db;
#pragma unroll
            for (int k = 0; k < KS; ++k)
                acc = fmaf(xe[li * KS + k][tid] * attl[li][k], dw[k], acc);
            accs[li] = acc;
            lsum += acc;
            lsq  = fmaf(acc, acc, lsq);
        }
        float4* ob4 = (float4*)obase;      // 16B aligned (r0 % 16 == 0)
#pragma unroll
        for (int v = 0; v < 4; ++v)
            ob4[v] = make_float4(accs[4 * v], accs[4 * v + 1],
                                 accs[4 * v + 2], accs[4 * v + 3]);
        __syncthreads();
    }

    atomicAdd(&stat_sum[tid], lsum);
    atomicAdd(&stat_sq[tid],  lsq);
}

// ---------------- kernel 2: finalize BN scale/shift per channel --------
__global__ void k_stats(float* ws, const float* __restrict__ gamma,
                        const float* __restrict__ beta) {
    int c = threadIdx.x;                   // 128 threads
    const float inv_n = 1.0f / (float)LTOT;
    float mean = ws[c] * inv_n;
    float var  = ws[128 + c] * inv_n - mean * mean;
    float sc   = rsqrtf(var + 1e-5f) * gamma[c];
    ws[256 + c] = sc;
    ws[384 + c] = beta[c] - mean * sc;
}

// ---------------- kernel 3: in-place normalize + SiLU (float4) ---------
__global__ __launch_bounds__(256) void k_norm(float4* __restrict__ out,
                                              const float* __restrict__ scale,
                                              const float* __restrict__ shift) {
    size_t i = (size_t)blockIdx.x * blockDim.x + threadIdx.x;
    // 20,971,520 floats / 4 = 5,242,880 float4; MM%4==0 so c uniform per vec
    int c = (int)((i >> 12) & (CCH - 1));  // (i*4 / MM) % C, MM/4 = 4096
    float sc = scale[c], sh = shift[c];
    float4 v = out[i];
    float y0 = v.x * sc + sh, y1 = v.y * sc + sh;
    float y2 = v.z * sc + sh, y3 = v.w * sc + sh;
    v.x = y0 * fast_sigmoid(y0);
    v.y = y1 * fast_sigmoid(y1);
    v.z = y2 * fast_sigmoid(y2);
    v.w = y3 * fast_sigmoid(y3);
    out[i] = v;
}

extern "C" void kernel_launch(void* const* d_in, const int* in_sizes, int n_in,
                              void* d_out, int out_size, void* d_ws, size_t ws_size,
                              hipStream_t stream) {
    const float* x       = (const float*)d_in[0];
    const int*   codes   = (const int*)  d_in[1];
    const float* att_w   = (const float*)d_in[2];
    const float* att_b   = (const float*)d_in[3];
    const float* depth_w = (const float*)d_in[4];
    const float* depth_b = (const float*)d_in[5];
    const float* gamma   = (const float*)d_in[6];
    const float* beta    = (const float*)d_in[7];
    float* out = (float*)d_out;
    float* ws  = (float*)d_ws;             // [0:128)=sum [128:256)=sumsq
                                           // [256:384)=scale [384:512)=shift
    k_init<<<1, 256, 0, stream>>>(ws);
    k_main<<<2560, 128, 0, stream>>>(x, codes, att_w, att_b, depth_w, depth_b,
                                     out, ws, ws + 128);
    k_stats<<<1, 128, 0, stream>>>(ws, gamma, beta);
    const int n4 = (LTOT * CCH) / 4;       // 5,242,880
    k_norm<<<n4 / 256, 256, 0, stream>>>((float4*)out, ws + 256, ws + 384);
}